// CantorMultiheadFusion_18829136626053
// MI455X (gfx1250) — compile-verified
//
#include <hip/hip_runtime.h>

// Problem constants (match reference).
#define B_DIM 2
#define S_DIM 4096
#define D_DIM 512
#define K_DIM 32

typedef __attribute__((ext_vector_type(2))) float v2f;
typedef __attribute__((ext_vector_type(8))) float v8f;

// GEMM tiling: block tile 128x64, 8 waves (wave32), each wave 32x32 (2x2 WMMA accs).
#define BM 128
#define BN 64
#define BK 32
#define LDA 36    // LDS row stride for A tile (floats); rows 16B-aligned, padded vs 32
#define LDBT 34   // LDS row stride for transposed B tile (floats); 8B-aligned, padded vs 32

// Async DMA global -> LDS, tracked by ASYNCcnt.
__device__ __forceinline__ void async_copy_b128(unsigned lds_addr, const void* gaddr) {
  asm volatile("global_load_async_to_lds_b128 %0, %1, off"
               :: "v"(lds_addr), "v"((unsigned long long)gaddr)
               : "memory");
}
__device__ __forceinline__ void async_copy_b32(unsigned lds_addr, const void* gaddr) {
  asm volatile("global_load_async_to_lds_b32 %0, %1, off"
               :: "v"(lds_addr), "v"((unsigned long long)gaddr)
               : "memory");
}

// C = A(MxK) * W(KxN) [+ bias + resid]  using V_WMMA_F32_16X16X4_F32 with
// double-buffered GLOBAL_LOAD_ASYNC_TO_LDS staging; B staged transposed so both
// A and B fragments are single ds_load_b64 ops (no register repacking).
template <bool EPILOGUE>
__global__ __launch_bounds__(256) void gemm_wmma_f32(
    const float* __restrict__ A,      // (M, Kk) row-major
    const float* __restrict__ W,      // (Kk, N) row-major
    const float* __restrict__ bias,   // (N) or nullptr
    const float* __restrict__ resid,  // (M, N) or nullptr
    float* __restrict__ C,            // (M, N)
    int M, int N, int Kk) {
  __shared__ float Ash[2][BM * LDA];   // A tile, row-major      (2 x 18KB)
  __shared__ float Bts[2][BN * LDBT];  // B tile, TRANSPOSED [col][k] (2 x 8.5KB)

  const int tid   = threadIdx.x;
  const int lane  = tid & 31;
  const int wave  = tid >> 5;   // 0..7
  const int wm    = wave & 3;   // wave row (M dir), 0..3
  const int wn    = wave >> 2;  // wave col (N dir), 0..1
  const int lhalf = lane >> 4;  // 0: lanes 0-15, 1: lanes 16-31
  const int lr    = lane & 15;

  const int m0 = blockIdx.x * BM;
  const int n0 = blockIdx.y * BN;

  v8f acc[2][2] = {};

  // Stage one K-chunk into LDS buffer `buf` via async DMA.
  // Per wave: 4 b128 ops (A, 128x32 row-major) + 8 b32 ops (B, 32x64 -> transposed)
  // = 12 outstanding ASYNC ops per chunk.
  auto stage = [&](int buf, int kc) {
#pragma unroll
    for (int i = 0; i < 4; ++i) {  // A: 1024 float4 total, 128 per wave
      int t = wave * 128 + i * 32 + lane;
      int r = t >> 3;
      int c = (t & 7) << 2;
      async_copy_b128((unsigned)(unsigned long long)&Ash[buf][r * LDA + c],
                      A + (size_t)(m0 + r) * Kk + kc * BK + c);
    }
#pragma unroll
    for (int i = 0; i < 8; ++i) {  // B: 2048 floats; lane->c contiguous global reads
      int t = tid + i * 256;
      int c = t & 63;              // column in tile (global-contiguous across lanes)
      int r = t >> 6;              // k within chunk
      async_copy_b32((unsigned)(unsigned long long)&Bts[buf][c * LDBT + r],
                     W + (size_t)(kc * BK + r) * N + n0 + c);
    }
  };

  const int nk = Kk / BK;
  stage(0, 0);  // prologue

  for (int kc = 0; kc < nk; ++kc) {
    const int cur = kc & 1;
    if (kc + 1 < nk) {
      stage(cur ^ 1, kc + 1);  // overlap next chunk's DMA with this chunk's WMMAs
      // 12 ops just issued for kc+1 may remain in flight; everything older (kc) done.
      asm volatile("s_wait_asynccnt 0xc" ::: "memory");
    } else {
      asm volatile("s_wait_asynccnt 0x0" ::: "memory");
    }
    __syncthreads();  // all waves' slices of chunk `cur` visible

#pragma unroll
    for (int kk = 0; kk < BK; kk += 4) {
      // Fragment layout (ISA): lane half 0 -> K=kk,kk+1 ; half 1 -> K=kk+2,kk+3.
      const int kb = kk + lhalf * 2;
      // A fragments: row-major LDS, one ds_load_b64 each.
      v2f a0 = *reinterpret_cast<const v2f*>(&Ash[cur][(wm * 32 + 0 + lr) * LDA + kb]);
      v2f a1 = *reinterpret_cast<const v2f*>(&Ash[cur][(wm * 32 + 16 + lr) * LDA + kb]);
      // B fragments: transposed LDS -> {B[kb][col], B[kb+1][col]} contiguous.
      const int c0 = wn * 32 + lr;
      v2f b0 = *reinterpret_cast<const v2f*>(&Bts[cur][(c0 + 0) * LDBT + kb]);
      v2f b1 = *reinterpret_cast<const v2f*>(&Bts[cur][(c0 + 16) * LDBT + kb]);
      // 8 args: (neg_a, A, neg_b, B, c_mod, C, reuse_a, reuse_b)
      acc[0][0] = __builtin_amdgcn_wmma_f32_16x16x4_f32(false, a0, false, b0,
                                                        (short)0, acc[0][0], false, false);
      acc[0][1] = __builtin_amdgcn_wmma_f32_16x16x4_f32(false, a0, false, b1,
                                                        (short)0, acc[0][1], false, false);
      acc[1][0] = __builtin_amdgcn_wmma_f32_16x16x4_f32(false, a1, false, b0,
                                                        (short)0, acc[1][0], false, false);
      acc[1][1] = __builtin_amdgcn_wmma_f32_16x16x4_f32(false, a1, false, b1,
                                                        (short)0, acc[1][1], false, false);
    }
    __syncthreads();  // nobody may re-stage `cur` until all waves finished reading it
  }

  // Epilogue. C/D layout: VGPR r -> M = r (lanes 0-15) / M = 8+r (lanes 16-31), N = lane&15.
#pragma unroll
  for (int mi = 0; mi < 2; ++mi) {
#pragma unroll
    for (int ni = 0; ni < 2; ++ni) {
      const int col = n0 + wn * 32 + ni * 16 + lr;
#pragma unroll
      for (int r = 0; r < 8; ++r) {
        const int row = m0 + wm * 32 + mi * 16 + lhalf * 8 + r;
        const size_t off = (size_t)row * N + col;
        float v = acc[mi][ni][r];
        if (EPILOGUE) v += bias[col] + resid[off];
        C[off] = v;
      }
    }
  }
}

// fused[b,s,:] = sum_k w[s,k] * h[b, routes[s,k], :]   (gather hits L2: h is 16MB total)
__global__ __launch_bounds__(256) void cantor_fuse_kernel(
    const float* __restrict__ h,       // (B, S, D)
    const float* __restrict__ w,       // (S, K)
    const int* __restrict__ routes,    // (S, K)
    float* __restrict__ fused) {       // (B, S, D)
  const int s = blockIdx.x;
  const int b = blockIdx.y;

  __shared__ int   r_sh[K_DIM];
  __shared__ float w_sh[K_DIM];
  if (threadIdx.x < K_DIM) {
    r_sh[threadIdx.x] = routes[(size_t)s * K_DIM + threadIdx.x];
    w_sh[threadIdx.x] = w[(size_t)s * K_DIM + threadIdx.x];
  }
  __syncthreads();

  const float* hb = h + (size_t)b * S_DIM * D_DIM;
  const int c0 = threadIdx.x;  // columns c0 and c0+256 (D=512, 256 threads)
  float acc0 = 0.f, acc1 = 0.f;
#pragma unroll 4
  for (int k = 0; k < K_DIM; ++k) {
    const float* row = hb + (size_t)r_sh[k] * D_DIM;
    const float wk = w_sh[k];
    acc0 = fmaf(wk, row[c0], acc0);
    acc1 = fmaf(wk, row[c0 + 256], acc1);
  }
  const size_t base = ((size_t)b * S_DIM + s) * D_DIM;
  fused[base + c0] = acc0;
  fused[base + c0 + 256] = acc1;
}

extern "C" void kernel_launch(void* const* d_in, const int* in_sizes, int n_in,
                              void* d_out, int out_size, void* d_ws, size_t ws_size,
                              hipStream_t stream) {
  const float* x      = (const float*)d_in[0];  // (B,S,D)
  const float* W_in   = (const float*)d_in[1];  // (D,D)
  const float* W_out  = (const float*)d_in[2];  // (D,D)
  const float* b_out  = (const float*)d_in[3];  // (D)
  const float* fw     = (const float*)d_in[4];  // (S,K)
  const int*   routes = (const int*)d_in[5];    // (S,K)
  float* out = (float*)d_out;

  float* h     = (float*)d_ws;                        // 16 MB
  float* fused = h + (size_t)B_DIM * S_DIM * D_DIM;   // next 16 MB

  const int M = B_DIM * S_DIM;  // 8192
  dim3 grid(M / BM, D_DIM / BN);  // 64 x 8

  // 1) h = x @ W_in
  gemm_wmma_f32<false><<<grid, 256, 0, stream>>>(x, W_in, nullptr, nullptr, h,
                                                 M, D_DIM, D_DIM);
  // 2) Cantor-routed weighted fusion
  cantor_fuse_kernel<<<dim3(S_DIM, B_DIM), 256, 0, stream>>>(h, fw, routes, fused);
  // 3) out = x + fused @ W_out + b_out
  gemm_wmma_f32<true><<<grid, 256, 0, stream>>>(fused, W_out, b_out, x, out,
                                                M, D_DIM, D_DIM);
}